// Repro_87806311399656
// MI455X (gfx1250) — compile-verified
//
#include <hip/hip_runtime.h>

// Problem sizes (fixed by the reference)
#define S_DIM 8192
#define K_DIM 2048
#define D_DIM 2048

#define TILE  128   // output macro-tile per block (M and N)
#define CHUNK 32    // reduction-dim chunk staged in LDS
#define NCH_A (S_DIM / CHUNK)
#define NCH_B (K_DIM / CHUNK)
#define LSTR  144   // reduction-major LDS row stride in floats (rows k,k+2 -> disjoint bank sets)
#define ASTR  36    // m-major LDS row stride (36*l mod 64 distinct for l<16; halves on banks 0/2 mod 4)

#if defined(__gfx1250__) && __has_builtin(__builtin_amdgcn_global_load_async_to_lds_b128) && \
    __has_builtin(__builtin_amdgcn_s_wait_asynccnt)
#define USE_ASYNC 1
#else
#define USE_ASYNC 0
#endif

typedef __attribute__((ext_vector_type(2))) float v2f;
typedef __attribute__((ext_vector_type(8))) float v8f;
typedef int v4i __attribute__((vector_size(4 * sizeof(int))));  // matches builtin's pointee type

// D = A(16x4) * B(4x16) + C, fp32 WMMA (CDNA5 V_WMMA_F32_16X16X4_F32)
__device__ __forceinline__ v8f wmma4(v2f a, v2f b, v8f c) {
  return __builtin_amdgcn_wmma_f32_16x16x4_f32(
      /*neg_a=*/false, a, /*neg_b=*/false, b,
      /*c_mod=*/(short)0, c, /*reuse_a=*/false, /*reuse_b=*/false);
}

// 16-byte global -> LDS copy: async (ASYNCcnt-tracked, no VGPR staging) when available.
__device__ __forceinline__ void cp16(float* lds_dst, const float* g_src) {
#if USE_ASYNC
  __builtin_amdgcn_global_load_async_to_lds_b128(
      (__attribute__((address_space(1))) v4i*)g_src,
      (__attribute__((address_space(3))) v4i*)lds_dst, 0, 0);
#else
  *(float4*)lds_dst = *(const float4*)g_src;
#endif
}

template <int N>
__device__ __forceinline__ void async_wait() {
#if USE_ASYNC
  __builtin_amdgcn_s_wait_asynccnt(N);
#endif
}

// ---------------------------------------------------------------------------
// Kernel 1: decayed[K,D] = exp(-a[k]*W[k,d]) * (Q^T @ (g ⊙ V))[k,d]
// Grid: (D/128, K/128), block: 256 threads = 8 waves (2x4 wave grid).
// ---------------------------------------------------------------------------
__global__ __launch_bounds__(256) void state_decay_kernel(
    const float* __restrict__ Q,       // [S,K]
    const float* __restrict__ g,       // [S]
    const float* __restrict__ V,       // [S,D]
    const float* __restrict__ aK,      // [K]
    const float* __restrict__ W,       // [K,D]
    float* __restrict__ decayed)       // [K,D]
{
  __shared__ float lds_a[2][CHUNK][LSTR]; // Q chunk, reduction-major: [s][k-col]
  __shared__ float lds_b[2][CHUNK][LSTR]; // raw V chunk, reduction-major: [s][d-col]
  __shared__ float lds_g[2][CHUNK];       // gate values for the chunk

  const int kTile = blockIdx.y * TILE;
  const int dTile = blockIdx.x * TILE;
  const int tid   = threadIdx.x;
  const int lane  = tid & 31;
  const int wave  = tid >> 5;
  const int mOff  = (wave >> 2) * 64;
  const int nOff  = (wave & 3) * 32;
  const int l15   = lane & 15;
  const int kp    = (lane >> 4) << 1;   // lanes 0-15 -> K 0/1, lanes 16-31 -> K 2/3

  // Stage one 32x128 chunk of Q and of V (4 async b128 per thread per array = 8/wave).
  auto issue = [&](int nb, int sBase) {
#pragma unroll
    for (int it = 0; it < 4; ++it) {
      const int idx = tid + it * 256;     // 0..1023
      const int row = idx >> 5;           // s within chunk, 0..31
      const int c4  = (idx & 31) << 2;    // col 0..124
      cp16(&lds_a[nb][row][c4], Q + (size_t)(sBase + row) * K_DIM + kTile + c4);
      cp16(&lds_b[nb][row][c4], V + (size_t)(sBase + row) * D_DIM + dTile + c4);
    }
    if (tid < CHUNK) lds_g[nb][tid] = g[sBase + tid];
  };

  v8f acc[4][2] = {};

  issue(0, 0);
  for (int c = 0; c < NCH_A; ++c) {
    const int cur = c & 1;
    if (c + 1 < NCH_A) {
      issue(cur ^ 1, (c + 1) * CHUNK);
      async_wait<8>();                   // retire current buffer's 8 (in-order), leave prefetch in flight
    } else {
      async_wait<0>();
    }
    __syncthreads();

#pragma unroll
    for (int kc = 0; kc < CHUNK; kc += 4) {
      const float g0 = lds_g[cur][kc + kp];       // broadcast within each 16-lane half
      const float g1 = lds_g[cur][kc + kp + 1];
      v2f af[4], bf[2];
#pragma unroll
      for (int mi = 0; mi < 4; ++mi) {
        af[mi].x = lds_a[cur][kc + kp    ][mOff + mi * 16 + l15];
        af[mi].y = lds_a[cur][kc + kp + 1][mOff + mi * 16 + l15];
      }
#pragma unroll
      for (int ni = 0; ni < 2; ++ni) {
        bf[ni].x = lds_b[cur][kc + kp    ][nOff + ni * 16 + l15] * g0;  // fold gate into B
        bf[ni].y = lds_b[cur][kc + kp + 1][nOff + ni * 16 + l15] * g1;
      }
#pragma unroll
      for (int mi = 0; mi < 4; ++mi)
#pragma unroll
        for (int ni = 0; ni < 2; ++ni)
          acc[mi][ni] = wmma4(af[mi], bf[ni], acc[mi][ni]);
    }
    __syncthreads();
  }

  // Fused decay epilogue; C layout: VGPR r -> row r (lanes<16) / r+8 (lanes>=16)
  const int mhalf = (lane >> 4) << 3;
#pragma unroll
  for (int mi = 0; mi < 4; ++mi) {
#pragma unroll
    for (int ni = 0; ni < 2; ++ni) {
      const int dcol = dTile + nOff + ni * 16 + l15;
#pragma unroll
      for (int r = 0; r < 8; ++r) {
        const int krow = kTile + mOff + mi * 16 + mhalf + r;
        const float w  = W[(size_t)krow * D_DIM + dcol];
        const float dec = __expf(-aK[krow] * w);
        decayed[(size_t)krow * D_DIM + dcol] = dec * acc[mi][ni][r];
      }
    }
  }
}

// ---------------------------------------------------------------------------
// Kernel 2: out[S,D] = Q[S,K] @ decayed[K,D]
// Grid: (D/128, S/128), same block/wave structure.
// ---------------------------------------------------------------------------
__global__ __launch_bounds__(256) void readout_kernel(
    const float* __restrict__ Q,        // [S,K]
    const float* __restrict__ decayed,  // [K,D]
    float* __restrict__ out)            // [S,D]
{
  __shared__ float lds_a[2][TILE][ASTR];  // Q chunk, m-major: [s-row][k] (direct row copy)
  __shared__ float lds_b[2][CHUNK][LSTR]; // decayed chunk, reduction-major: [k][d-col]

  const int sTile = blockIdx.y * TILE;
  const int dTile = blockIdx.x * TILE;
  const int tid   = threadIdx.x;
  const int lane  = tid & 31;
  const int wave  = tid >> 5;
  const int mOff  = (wave >> 2) * 64;
  const int nOff  = (wave & 3) * 32;
  const int l15   = lane & 15;
  const int kp    = (lane >> 4) << 1;

  auto issue = [&](int nb, int kBase) {
#pragma unroll
    for (int it = 0; it < 4; ++it) {
      const int idx = tid + it * 256;     // 0..1023
      // A: 128 rows x 32 k, straight row copy into m-major layout
      const int m  = idx >> 3;            // 0..127
      const int a4 = (idx & 7) << 2;      // 0..28
      cp16(&lds_a[nb][m][a4], Q + (size_t)(sTile + m) * K_DIM + kBase + a4);
      // B: 32 rows x 128 d, straight row copy
      const int row = idx >> 5;           // 0..31
      const int b4  = (idx & 31) << 2;    // 0..124
      cp16(&lds_b[nb][row][b4], decayed + (size_t)(kBase + row) * D_DIM + dTile + b4);
    }
  };

  v8f acc[4][2] = {};

  issue(0, 0);
  for (int c = 0; c < NCH_B; ++c) {
    const int cur = c & 1;
    if (c + 1 < NCH_B) {
      issue(cur ^ 1, (c + 1) * CHUNK);
      async_wait<8>();
    } else {
      async_wait<0>();
    }
    __syncthreads();

#pragma unroll
    for (int kc = 0; kc < CHUNK; kc += 4) {
      v2f af[4], bf[2];
#pragma unroll
      for (int mi = 0; mi < 4; ++mi) {
        af[mi].x = lds_a[cur][mOff + mi * 16 + l15][kc + kp    ];
        af[mi].y = lds_a[cur][mOff + mi * 16 + l15][kc + kp + 1];
      }
#pragma unroll
      for (int ni = 0; ni < 2; ++ni) {
        bf[ni].x = lds_b[cur][kc + kp    ][nOff + ni * 16 + l15];
        bf[ni].y = lds_b[cur][kc + kp + 1][nOff + ni * 16 + l15];
      }
#pragma unroll
      for (int mi = 0; mi < 4; ++mi)
#pragma unroll
        for (int ni = 0; ni < 2; ++ni)
          acc[mi][ni] = wmma4(af[mi], bf[ni], acc[mi][ni]);
    }
    __syncthreads();
  }

  const int mhalf = (lane >> 4) << 3;
#pragma unroll
  for (int mi = 0; mi < 4; ++mi) {
#pragma unroll
    for (int ni = 0; ni < 2; ++ni) {
      const int dcol = dTile + nOff + ni * 16 + l15;
#pragma unroll
      for (int r = 0; r < 8; ++r) {
        const int srow = sTile + mOff + mi * 16 + mhalf + r;
        out[(size_t)srow * D_DIM + dcol] = acc[mi][ni][r];
      }
    }
  }
}

extern "C" void kernel_launch(void* const* d_in, const int* in_sizes, int n_in,
                              void* d_out, int out_size, void* d_ws, size_t ws_size,
                              hipStream_t stream) {
  const float* Q  = (const float*)d_in[0];  // [S,K]
  const float* g  = (const float*)d_in[1];  // [S]
  const float* V  = (const float*)d_in[2];  // [S,D]
  const float* aK = (const float*)d_in[3];  // [K]
  const float* W  = (const float*)d_in[4];  // [K,D]

  float* out     = (float*)d_out;                    // [S,D]
  float* decayed = out + (size_t)S_DIM * D_DIM;      // [K,D] (second tuple output)

  dim3 blk(256, 1, 1);
  dim3 gridA(D_DIM / TILE, K_DIM / TILE, 1);   // 16 x 16
  state_decay_kernel<<<gridA, blk, 0, stream>>>(Q, g, V, aK, W, decayed);

  dim3 gridB(D_DIM / TILE, S_DIM / TILE, 1);   // 16 x 64
  readout_kernel<<<gridB, blk, 0, stream>>>(Q, decayed, out);
}